// DurationCalculator_49383533969826
// MI455X (gfx1250) — compile-verified
//
#include <hip/hip_runtime.h>
#include <stdint.h>

// Problem constants (reference: B=32, H=24, L=800, T=200)
#define B_ 32
#define H_ 24
#define L_ 800
#define T_ 200

#define TILE_ROWS 16                        // rows per tile
#define TILE_FLOATS (TILE_ROWS * T_)        // 3200 floats
#define TILE_BYTES  (TILE_FLOATS * 4)       // 12800 bytes
#define NTILES      (L_ / TILE_ROWS)        // 50 tiles per (b,h)
#define SLICE_ELEMS (2 * T_)                // 400 floats: one wave's 2 rows
#define SLICE_BYTES (SLICE_ELEMS * 4)       // 1600 bytes

typedef uint32_t v4u __attribute__((ext_vector_type(4)));
typedef uint32_t v8u __attribute__((ext_vector_type(8)));

// ---- CDNA5 per-lane async global->LDS copy (ASYNCcnt-tracked) ----
__device__ __forceinline__ void async_ld_b128(uint32_t lds_byte_addr,
                                              uint32_t gbyte_off,
                                              const char* sbase) {
  asm volatile("global_load_async_to_lds_b128 %0, %1, %2"
               :: "v"(lds_byte_addr), "v"(gbyte_off), "s"(sbase)
               : "memory");
}

// ---- CDNA5 Tensor Data Mover: one DMA per wave-slice (TENSORcnt-tracked) ----
// 1-D tile: 400 x 4-byte elements, contiguous in global and in LDS.
__device__ __forceinline__ void tdm_load_slice(uint64_t gaddr, uint32_t lds_addr) {
  v4u g0;
  g0[0] = 1u;                                            // count=1 (valid D#)
  g0[1] = lds_addr;                                      // lds_addr[31:0]
  g0[2] = (uint32_t)gaddr;                               // global_addr[31:0]
  g0[3] = ((uint32_t)(gaddr >> 32) & 0x01FFFFFFu)        // global_addr[56:32]
          | (2u << 30);                                  // type=2 ("image")
  v8u g1;
  g1[0] = 2u << 16;                                      // data_size=4B
  g1[1] = (uint32_t)SLICE_ELEMS << 16;                   // tensor_dim0[15:0]
  g1[2] = 1u << 16;                                      // dim0 hi=0, tensor_dim1=1
  g1[3] = (uint32_t)SLICE_ELEMS << 16;                   // tile_dim0=400
  g1[4] = 0u;                                            // tile_dim1/2 unused
  g1[5] = (uint32_t)SLICE_ELEMS;                         // tensor_dim0_stride
  g1[6] = 0u;
  g1[7] = 0u;
  asm volatile("tensor_load_to_lds %0, %1" :: "s"(g0), "s"(g1) : "memory");
}

// ---------------- Phase 0: zero the (B,T) int32 output ----------------
__global__ void zero_out_kernel(int* __restrict__ out, int n) {
  int i = blockIdx.x * blockDim.x + threadIdx.x;
  if (i < n) out[i] = 0;
}

// ---------------- Phase 1: per-(b,h) sum of row maxes ----------------
// One block per (b,h), 8 waves. Each wave owns a private 2-row slice of every
// 16-row tile: even tiles prefetched by the TDM (TENSORcnt), odd tiles by
// per-lane async B128 copies (ASYNCcnt). Private slices + split counters
// => zero per-tile barriers, full copy/compute overlap.
__global__ void __launch_bounds__(256)
phase1_rowmax_kernel(const float* __restrict__ att, float* __restrict__ partial) {
  __shared__ float sbuf[2][TILE_FLOATS];   // 2 x 12.8 KB double buffer
  __shared__ float wsum[8];

  const int lane = threadIdx.x & 31;
  const int wids = __builtin_amdgcn_readfirstlane((int)(threadIdx.x >> 5));
  const int bh   = blockIdx.x;             // bh = b*H + h

  const char* gbase = (const char*)(att + (size_t)bh * (size_t)(L_ * T_));
  const uint32_t lb0 = (uint32_t)(size_t)(void*)&sbuf[0][0];
  const uint32_t lb1 = (uint32_t)(size_t)(void*)&sbuf[1][0];
  const uint32_t sliceLds = (uint32_t)(wids * SLICE_BYTES);

  // prime: tile 0 via TDM
  tdm_load_slice((uint64_t)(uintptr_t)(gbase + (size_t)wids * SLICE_BYTES),
                 lb0 + sliceLds);

  float acc = 0.0f;
  for (int t = 0; t < NTILES; ++t) {
    const int nt = t + 1;
    if (nt < NTILES) {
      const char* gslice =
          gbase + (size_t)nt * TILE_BYTES + (size_t)wids * SLICE_BYTES;
      const uint32_t dst = ((nt & 1) ? lb1 : lb0) + sliceLds;
      if (nt & 1) {
        // 100 x 16B chunks over 32 lanes: 3 full strides + 4-lane tail
        uint32_t c = (uint32_t)lane;
        async_ld_b128(dst + c * 16u,        c * 16u,        gslice);
        async_ld_b128(dst + (c + 32) * 16u, (c + 32) * 16u, gslice);
        async_ld_b128(dst + (c + 64) * 16u, (c + 64) * 16u, gslice);
        if (lane < 4)
          async_ld_b128(dst + (c + 96) * 16u, (c + 96) * 16u, gslice);
      } else {
        tdm_load_slice((uint64_t)(uintptr_t)gslice, dst);
      }
    }
    // tile t sits on the *other* counter than the prefetch just issued, so a
    // wait-to-zero here never stalls the in-flight tile t+1 transfer.
    if (t & 1) asm volatile("s_wait_asynccnt 0x0" ::: "memory");
    else       asm volatile("s_wait_tensorcnt 0x0" ::: "memory");

    const float* rows = &sbuf[t & 1][wids * SLICE_ELEMS];
#pragma unroll
    for (int rr = 0; rr < 2; ++rr) {
      const float* row = rows + rr * T_;
      float m = -3.402823466e38f;
      for (int k = lane; k < T_; k += 32) m = fmaxf(m, row[k]);   // ds loads
#pragma unroll
      for (int off = 16; off > 0; off >>= 1)
        m = fmaxf(m, __shfl_xor(m, off, 32));                     // wave32
      acc += m;                              // uniform across the wave
    }
  }

  if (lane == 0) wsum[wids] = acc;
  __syncthreads();
  if (threadIdx.x == 0) {
    float s = 0.0f;
#pragma unroll
    for (int i = 0; i < 8; ++i) s += wsum[i];
    partial[bh] = s;                         // mean scaling is monotonic: skip
  }
}

// ---------------- Phase 2: argmax head over summed partials ----------------
__global__ void pick_head_kernel(const float* __restrict__ partial,
                                 int* __restrict__ headp) {
  int lane = threadIdx.x;                    // 32 threads, 1 wave
  float s = -3.402823466e38f;
  if (lane < H_) {
    float a = 0.0f;
    for (int b = 0; b < B_; ++b) a += partial[b * H_ + lane];
    s = a;
  }
  int idx = lane;
#pragma unroll
  for (int off = 16; off > 0; off >>= 1) {
    float ov = __shfl_xor(s, off, 32);
    int   oi = __shfl_xor(idx, off, 32);
    if (ov > s || (ov == s && oi < idx)) { s = ov; idx = oi; }
  }
  if (lane == 0) *headp = idx;
}

// ---------------- Phase 3: masked argmax per (b,l) + histogram ----------------
// One wave per output row; the winning head's 20.5 MB slice is L2-resident
// (192 MB L2) after the phase-1 stream.
__global__ void __launch_bounds__(256)
durations_kernel(const float* __restrict__ att, const int* __restrict__ ilens,
                 const int* __restrict__ olens, const int* __restrict__ headp,
                 int* __restrict__ out) {
  const int wid  = threadIdx.x >> 5;
  const int lane = threadIdx.x & 31;
  const int row  = blockIdx.x * 8 + wid;     // [0, B*L)
  const int b = row / L_;
  const int l = row - b * L_;
  const int head = *headp;                   // uniform scalar load
  const int ilen = ilens[b];

  const float4* r4 =
      (const float4*)(att + ((size_t)(b * H_ + head) * L_ + l) * (size_t)T_);

  float bv = -3.402823466e38f;
  int   bi = 0;
  for (int c = lane; c < T_ / 4; c += 32) {  // per-lane t ascending -> first-max
    float4 v = r4[c];
    int t = 4 * c;
    if (t     < ilen && v.x > bv) { bv = v.x; bi = t;     }
    if (t + 1 < ilen && v.y > bv) { bv = v.y; bi = t + 1; }
    if (t + 2 < ilen && v.z > bv) { bv = v.z; bi = t + 2; }
    if (t + 3 < ilen && v.w > bv) { bv = v.w; bi = t + 3; }
  }
  // cross-lane argmax with first-occurrence (smallest index) tie-break
#pragma unroll
  for (int off = 16; off > 0; off >>= 1) {
    float ov = __shfl_xor(bv, off, 32);
    int   oi = __shfl_xor(bi, off, 32);
    if (ov > bv || (ov == bv && oi < bi)) { bv = ov; bi = oi; }
  }
  if (lane == 0 && l < olens[b])
    atomicAdd(&out[b * T_ + bi], 1);
}

extern "C" void kernel_launch(void* const* d_in, const int* in_sizes, int n_in,
                              void* d_out, int out_size, void* d_ws, size_t ws_size,
                              hipStream_t stream) {
  const float* att   = (const float*)d_in[0];   // (B,H,L,T) f32
  const int*   ilens = (const int*)d_in[1];     // (B,) i32
  const int*   olens = (const int*)d_in[2];     // (B,) i32
  int* out = (int*)d_out;                       // (B,T) int32 durations

  float* partial = (float*)d_ws;                        // B*H = 768 floats
  int*   headp   = (int*)((char*)d_ws + B_ * H_ * 4);   // 1 int

  zero_out_kernel<<<(B_ * T_ + 255) / 256, 256, 0, stream>>>(out, B_ * T_);
  phase1_rowmax_kernel<<<B_ * H_, 256, 0, stream>>>(att, partial);
  pick_head_kernel<<<1, 32, 0, stream>>>(partial, headp);
  durations_kernel<<<(B_ * L_) / 8, 256, 0, stream>>>(att, ilens, olens, headp, out);
}